// Filter_33062658244750
// MI455X (gfx1250) — compile-verified
//
#include <hip/hip_runtime.h>

#define BATCH 4096
#define SEQ   512
#define STATE 16
#define OBSN  8
#define INPN  4

typedef float v2f __attribute__((ext_vector_type(2)));
typedef float v4f __attribute__((ext_vector_type(4)));
typedef float v8f __attribute__((ext_vector_type(8)));

static __device__ __forceinline__ v8f wmma4(v2f a, v2f b, v8f c) {
  // D(16x16,f32) = A(16x4,f32) * B(4x16,f32) + C ; full precision fp32 WMMA
  return __builtin_amdgcn_wmma_f32_16x16x4_f32(false, a, false, b, (short)0, c,
                                               false, false);
}

// ---------------------------------------------------------------------------
// Phase 1: batch-independent Riccati recursion. One workgroup, sequential in t.
// Produces M_t = (I-K_t C)A, K_t, G_t = (I-K_t C)B for all 512 steps.
// ---------------------------------------------------------------------------
__global__ __launch_bounds__(256) void riccati_kernel(
    const float* __restrict__ A, const float* __restrict__ Bm,
    const float* __restrict__ C, const float* __restrict__ Q,
    const float* __restrict__ R,
    float* __restrict__ Mout,   // SEQ x 16 x 16
    float* __restrict__ Kout,   // SEQ x 16 x 8
    float* __restrict__ Gout)   // SEQ x 16 x 4
{
  __shared__ float sA[256], sB[64], sC[128], sQ[256], sR[64];
  __shared__ float P[256], T1[256], Pp[256], T2[128], Saug[128], Kk[128], IKC[256];
  const int tid = threadIdx.x;
  const int i = tid >> 4, j = tid & 15;

  sA[tid] = A[tid];
  sQ[tid] = Q[tid];
  if (tid < 64)  { sB[tid] = Bm[tid]; sR[tid] = R[tid]; }
  if (tid < 128) { sC[tid] = C[tid]; }
  P[tid] = (i == j) ? 1.0f : 0.0f;   // P0 = I
  __syncthreads();

  for (int t = 0; t < SEQ; ++t) {
    // T1 = P * A^T
    {
      float s = 0.f;
      for (int k = 0; k < 16; ++k) s += P[i*16+k] * sA[j*16+k];
      T1[tid] = s;
    }
    __syncthreads();
    // Pp = A * T1 + Q
    {
      float s = sQ[tid];
      for (int k = 0; k < 16; ++k) s += sA[i*16+k] * T1[k*16+j];
      Pp[tid] = s;
    }
    __syncthreads();
    // T2 = Pp * C^T  (16x8)
    if (j < 8) {
      float s = 0.f;
      for (int k = 0; k < 16; ++k) s += Pp[i*16+k] * sC[j*16+k];
      T2[i*8+j] = s;
    }
    __syncthreads();
    // Saug = [ C*T2 + R | I ]   (8 x 16 augmented)
    if (tid < 128) {
      int si = tid >> 4, sj = tid & 15;
      float v;
      if (sj < 8) {
        float s = sR[si*8+sj];
        for (int k = 0; k < 16; ++k) s += sC[si*16+k] * T2[k*8+sj];
        v = s;
      } else {
        v = ((sj - 8) == si) ? 1.0f : 0.0f;
      }
      Saug[tid] = v;
    }
    __syncthreads();
    // Gauss-Jordan inverse of S (S = R + PSD: no pivoting needed)
    for (int c = 0; c < 8; ++c) {
      if (tid < 16) {
        float piv = Saug[c*16 + c];          // all 16 lanes read before any write
        Saug[c*16 + tid] = Saug[c*16 + tid] / piv;
      }
      __syncthreads();
      float f = 0.f;
      if (tid < 128) f = Saug[(tid >> 4)*16 + c];
      __syncthreads();
      if (tid < 128) {
        int si = tid >> 4, sj = tid & 15;
        if (si != c) Saug[tid] = Saug[tid] - f * Saug[c*16 + sj];
      }
      __syncthreads();
    }
    // Kk = T2 * Sinv  (16x8)
    if (tid < 128) {
      int ki = tid >> 3, kj = tid & 7;
      float s = 0.f;
      for (int k = 0; k < 8; ++k) s += T2[ki*8+k] * Saug[k*16 + 8 + kj];
      Kk[tid] = s;
    }
    __syncthreads();
    // IKC = I - Kk*C
    {
      float s = (i == j) ? 1.0f : 0.0f;
      for (int k = 0; k < 8; ++k) s -= Kk[i*8+k] * sC[k*16+j];
      IKC[tid] = s;
    }
    __syncthreads();
    // M_t = IKC*A ; P = IKC*Pp ; G_t = IKC*B ; emit K_t
    {
      float m = 0.f, p = 0.f;
      for (int k = 0; k < 16; ++k) {
        m += IKC[i*16+k] * sA[k*16+j];
        p += IKC[i*16+k] * Pp[k*16+j];
      }
      Mout[t*256 + tid] = m;
      P[tid] = p;
    }
    if (tid < 128) Kout[t*128 + tid] = Kk[tid];
    if (tid < 64) {
      int gi = tid >> 2, gj = tid & 3;
      float g = 0.f;
      for (int k = 0; k < 16; ++k) g += IKC[gi*16+k] * sB[k*4+gj];
      Gout[t*64 + tid] = g;
    }
    __syncthreads();
  }
}

// ---------------------------------------------------------------------------
// Phase 2: streaming state recursion. One wave owns TPW tiles of 16 batches.
// x_t (16 state x 16 batch) = M_t*x_{t-1} + G_t*u_t + K_t*z_t via 7 chained
// v_wmma_f32_16x16x4_f32 per tile per step. X round-trips through wave-private
// LDS to convert D-layout -> B-operand layout (same-wave LDS ops are in-order).
// ---------------------------------------------------------------------------
#define TPW   2   // 16-batch tiles per wave (ILP across the serial time chain)
#define WAVES 2   // waves per block

__global__ __launch_bounds__(WAVES*32) void kalman_stream_kernel(
    const float* __restrict__ obs, const float* __restrict__ u,
    const float* __restrict__ x0,
    const float* __restrict__ Mw, const float* __restrict__ Kw,
    const float* __restrict__ Gw,
    float* __restrict__ out)
{
  __shared__ float xs_all[WAVES][TPW][256];
  const int lane = threadIdx.x & 31;
  const int wave = threadIdx.x >> 5;
  const int half = lane >> 4;     // 0: lanes 0-15, 1: lanes 16-31
  const int col  = lane & 15;     // N (batch) column / M (row) for A-operand
  const int tile0 = (blockIdx.x * WAVES + wave) * TPW;

  float* xs[TPW];
  int base[TPW];
  for (int tt = 0; tt < TPW; ++tt) {
    xs[tt]   = &xs_all[wave][tt][0];
    base[tt] = (tile0 + tt) * 16;
    // X_0 = x0 broadcast across batch columns (LDS layout: xs[state*16 + batch])
    for (int r = 0; r < 8; ++r) {
      int s = r + 8*half;
      xs[tt][s*16 + col] = x0[s];
    }
  }

  for (int t = 0; t < SEQ; ++t) {
    const float* Mt = Mw + t*256;
    const float* Kt = Kw + t*128;
    const float* Gt = Gw + t*64;
    // A-operands, shared by all tiles (hot in L2 across the whole GPU).
    // A layout: lane holds row M=col; VGPR j holds K = chunk + j + 2*half.
    v2f aM[4], aK[2], aG;
    for (int kc = 0; kc < 4; ++kc)
      aM[kc] = *(const v2f*)(Mt + col*16 + 4*kc + 2*half);
    for (int kc = 0; kc < 2; ++kc)
      aK[kc] = *(const v2f*)(Kt + col*8  + 4*kc + 2*half);
    aG = *(const v2f*)(Gt + col*4 + 2*half);

    v8f acc[TPW];
    for (int tt = 0; tt < TPW; ++tt) {
      const int b = base[tt] + col;
      const size_t bt = (size_t)b * SEQ + t;
      // B-operands: row K = chunk + j + 2*half, column N = batch. Streamed
      // read-once data -> non-temporal loads.
      v2f zb0 = __builtin_nontemporal_load((const v2f*)(obs + bt*OBSN + 2*half));
      v2f zb1 = __builtin_nontemporal_load((const v2f*)(obs + bt*OBSN + 4 + 2*half));
      v2f ub  = __builtin_nontemporal_load((const v2f*)(u   + bt*INPN + 2*half));
      v2f xb0, xb1, xb2, xb3;
      xb0.x = xs[tt][(0  + 2*half)*16 + col]; xb0.y = xs[tt][(1  + 2*half)*16 + col];
      xb1.x = xs[tt][(4  + 2*half)*16 + col]; xb1.y = xs[tt][(5  + 2*half)*16 + col];
      xb2.x = xs[tt][(8  + 2*half)*16 + col]; xb2.y = xs[tt][(9  + 2*half)*16 + col];
      xb3.x = xs[tt][(12 + 2*half)*16 + col]; xb3.y = xs[tt][(13 + 2*half)*16 + col];

      v8f c = {};
      c = wmma4(aG,    ub,  c);   // + G_t * u_t
      c = wmma4(aK[0], zb0, c);   // + K_t * z_t   (K = 0..3)
      c = wmma4(aK[1], zb1, c);   //               (K = 4..7)
      c = wmma4(aM[0], xb0, c);   // + M_t * x     (K = 0..3)
      c = wmma4(aM[1], xb1, c);
      c = wmma4(aM[2], xb2, c);
      c = wmma4(aM[3], xb3, c);
      acc[tt] = c;
    }
    // Write back: LDS (next step's B operand) + global (write-once -> NT b128s)
    for (int tt = 0; tt < TPW; ++tt) {
      const int b = base[tt] + col;
      const size_t ooff = ((size_t)b * SEQ + t) * STATE + 8*half;
      for (int r = 0; r < 8; ++r)
        xs[tt][(r + 8*half)*16 + col] = acc[tt][r];
      v4f lo = { acc[tt][0], acc[tt][1], acc[tt][2], acc[tt][3] };
      v4f hi = { acc[tt][4], acc[tt][5], acc[tt][6], acc[tt][7] };
      __builtin_nontemporal_store(lo, (v4f*)(out + ooff));
      __builtin_nontemporal_store(hi, (v4f*)(out + ooff + 4));
    }
  }
}

// ---------------------------------------------------------------------------
extern "C" void kernel_launch(void* const* d_in, const int* in_sizes, int n_in,
                              void* d_out, int out_size, void* d_ws, size_t ws_size,
                              hipStream_t stream) {
  const float* obs = (const float*)d_in[0];  // (4096, 512, 8)
  const float* u   = (const float*)d_in[1];  // (4096, 512, 4)
  const float* A   = (const float*)d_in[2];  // (16, 16)
  const float* B   = (const float*)d_in[3];  // (16, 4)
  const float* C   = (const float*)d_in[4];  // (8, 16)
  const float* Q   = (const float*)d_in[5];  // (16, 16)
  const float* R   = (const float*)d_in[6];  // (8, 8)
  const float* x0  = (const float*)d_in[7];  // (16,)
  float* out = (float*)d_out;                // (4096, 512, 16)

  float* wsM = (float*)d_ws;                 // SEQ*256 floats
  float* wsK = wsM + (size_t)SEQ * 256;      // SEQ*128 floats
  float* wsG = wsK + (size_t)SEQ * 128;      // SEQ*64  floats

  riccati_kernel<<<1, 256, 0, stream>>>(A, B, C, Q, R, wsM, wsK, wsG);

  const int batches_per_block = WAVES * TPW * 16;   // 64
  kalman_stream_kernel<<<BATCH / batches_per_block, WAVES * 32, 0, stream>>>(
      obs, u, x0, wsM, wsK, wsG, out);
}